// Classify_Anything_Loss_19344532701384
// MI455X (gfx1250) — compile-verified
//
#include <hip/hip_runtime.h>

// Problem constants (from reference)
#define NROW 4096        // 2 * BATCH_SIZE
#define FEAT 300
#define BATCH 2048
#define SINK_ITERS 25
// TEMPERATURE = epsilon = 0.1 -> 1/T = 10

typedef __attribute__((ext_vector_type(2))) float v2f;
typedef __attribute__((ext_vector_type(8))) float v8f;

__device__ __forceinline__ v8f wmma_f32_16x16x4(v2f a, v2f b, v8f c) {
    // 8 args: (neg_a, A, neg_b, B, c_mod, C, reuse_a, reuse_b)
    return __builtin_amdgcn_wmma_f32_16x16x4_f32(
        false, a, false, b, (short)0, c, false, false);
}

// ---------------------------------------------------------------------------
// K[i][j] = (i==j) ? 0 : exp((exp(dot(F_i, L_j)*10) - 1)*10)
// 128x128 tile per block; 8 waves; each wave computes a 32x64 subtile
// (2 m-tiles x 4 n-tiles of 16x16) via V_WMMA_F32_16X16X4_F32.
// ---------------------------------------------------------------------------
__global__ __launch_bounds__(256) void gemm_k_kernel(
    const float* __restrict__ F, const float* __restrict__ L,
    float* __restrict__ K)
{
    const int lane = threadIdx.x & 31;
    const int wave = threadIdx.x >> 5;
    const int wr   = wave >> 1;             // 0..3  (row group)
    const int wc   = wave & 1;              // 0..1  (col group)
    const int rbase = blockIdx.y * 128 + wr * 32;
    const int cbase = blockIdx.x * 128 + wc * 64;

    // fragment addressing for 16x16x4 f32 WMMA:
    //   A[m][k]: lane = (k/2)*16 + m, vgpr = k%2  -> lane holds 2 consecutive k
    //   B[k][n]: lane = (k/2)*16 + n, vgpr = k%2  (B = L^T, so read L row n)
    const int lm = lane & 15;               // m (or n) within tile
    const int lk = (lane >> 4) << 1;        // k offset 0 or 2

    const float* a0 = F + (size_t)(rbase + lm) * FEAT + lk;        // m-tile 0
    const float* a1 = a0 + (size_t)16 * FEAT;                      // m-tile 1
    const float* b0 = L + (size_t)(cbase + lm) * FEAT + lk;        // n-tile 0

    v8f acc[2][4];
#pragma unroll
    for (int mt = 0; mt < 2; ++mt)
#pragma unroll
        for (int nt = 0; nt < 4; ++nt)
            acc[mt][nt] = (v8f)(0.0f);

    for (int k0 = 0; k0 < FEAT; k0 += 4) {
        v2f A0 = *(const v2f*)(a0 + k0);
        v2f A1 = *(const v2f*)(a1 + k0);
#pragma unroll
        for (int nt = 0; nt < 4; ++nt) {
            v2f B = *(const v2f*)(b0 + (size_t)nt * 16 * FEAT + k0);
            acc[0][nt] = wmma_f32_16x16x4(A0, B, acc[0][nt]);
            acc[1][nt] = wmma_f32_16x16x4(A1, B, acc[1][nt]);
        }
    }

    // C/D layout: vgpr e holds row (e + 8*(lane>=16)), col = lane%16
    const int hi8 = (lane >> 4) << 3;
#pragma unroll
    for (int mt = 0; mt < 2; ++mt) {
        const int row0 = rbase + mt * 16 + hi8;
#pragma unroll
        for (int nt = 0; nt < 4; ++nt) {
            const int col = cbase + nt * 16 + lm;
#pragma unroll
            for (int e = 0; e < 8; ++e) {
                const int row = row0 + e;
                float dot = acc[mt][nt][e];
                float sim = __expf(dot * 10.0f);          // exp(dot / T)
                float kv  = __expf((sim - 1.0f) * 10.0f); // exp(-cost / eps)
                if (row == col) kv = 0.0f;                // diagonal mask
                K[(size_t)row * NROW + col] = kv;
            }
        }
    }
}

// ---------------------------------------------------------------------------
// v[j] = 1
// ---------------------------------------------------------------------------
__global__ void init_ones_kernel(float* __restrict__ v)
{
    v[blockIdx.x * 256 + threadIdx.x] = 1.0f;
}

// ---------------------------------------------------------------------------
// y[row] = 1 / sum_j K[row][j] * x[j]   (row normalization scale)
// ---------------------------------------------------------------------------
__global__ __launch_bounds__(256) void rowscale_kernel(
    const float* __restrict__ K, const float* __restrict__ x,
    float* __restrict__ y)
{
    __shared__ float red[256];
    const int row = blockIdx.x;
    const float* Kr = K + (size_t)row * NROW;
    float acc = 0.0f;
    for (int j = threadIdx.x; j < NROW; j += 256) {
        __builtin_prefetch(Kr + j + 1024, 0, 0);  // global_prefetch_b8
        acc += Kr[j] * x[j];
    }
    red[threadIdx.x] = acc;
    __syncthreads();
#pragma unroll
    for (int s = 128; s > 0; s >>= 1) {
        if (threadIdx.x < s) red[threadIdx.x] += red[threadIdx.x + s];
        __syncthreads();
    }
    if (threadIdx.x == 0) y[row] = 1.0f / red[0];
}

// ---------------------------------------------------------------------------
// part[by][col] = sum_{r in 256-row slice} K[r][col] * x[r]
// 64 columns per block, coalesced across lanes; 4-way LDS reduce.
// ---------------------------------------------------------------------------
__global__ __launch_bounds__(256) void colpart_kernel(
    const float* __restrict__ K, const float* __restrict__ x,
    float* __restrict__ part)
{
    __shared__ float red[256];
    const int tx  = threadIdx.x & 63;
    const int ty  = threadIdx.x >> 6;   // 0..3
    const int col = blockIdx.x * 64 + tx;
    const int r0  = blockIdx.y * 256;
    float acc = 0.0f;
    for (int r = r0 + ty; r < r0 + 256; r += 4)
        acc += K[(size_t)r * NROW + col] * x[r];
    red[threadIdx.x] = acc;
    __syncthreads();
    if (ty == 0)
        part[(size_t)blockIdx.y * NROW + col] =
            red[tx] + red[tx + 64] + red[tx + 128] + red[tx + 192];
}

// ---------------------------------------------------------------------------
// y[j] = 1 / sum_b part[b][j]   (column normalization scale)
// ---------------------------------------------------------------------------
__global__ void colfinish_kernel(const float* __restrict__ part,
                                 float* __restrict__ y)
{
    const int j = blockIdx.x * 256 + threadIdx.x;
    float s = 0.0f;
#pragma unroll
    for (int b = 0; b < 16; ++b) s += part[(size_t)b * NROW + j];
    y[j] = 1.0f / s;
}

// ---------------------------------------------------------------------------
// loss = -(1/N) * sum_t log(u[t] * K[t][t^BATCH] * v[t^BATCH])
// (p = diag(u) K diag(v); the +B/-B diagonal log-sum is transpose-invariant)
// ---------------------------------------------------------------------------
__global__ __launch_bounds__(256) void loss_kernel(
    const float* __restrict__ K, const float* __restrict__ u,
    const float* __restrict__ v, float* __restrict__ out)
{
    __shared__ float red[256];
    float acc = 0.0f;
    for (int t = threadIdx.x; t < NROW; t += 256) {
        const int j = t ^ BATCH;
        float p = u[t] * K[(size_t)t * NROW + j] * v[j];
        acc += logf(p);
    }
    red[threadIdx.x] = acc;
    __syncthreads();
#pragma unroll
    for (int s = 128; s > 0; s >>= 1) {
        if (threadIdx.x < s) red[threadIdx.x] += red[threadIdx.x + s];
        __syncthreads();
    }
    if (threadIdx.x == 0) out[0] = -red[0] / (float)NROW;
}

// ---------------------------------------------------------------------------
extern "C" void kernel_launch(void* const* d_in, const int* in_sizes, int n_in,
                              void* d_out, int out_size, void* d_ws, size_t ws_size,
                              hipStream_t stream)
{
    const float* F = (const float*)d_in[0];   // features     [4096, 300] f32
    const float* L = (const float*)d_in[1];   // labels_vector[4096, 300] f32
    float* out = (float*)d_out;               // scalar f32 loss

    // workspace: K (64 MiB) | u (4096) | v (4096) | column partials (16*4096)
    float* Kw   = (float*)d_ws;
    float* u    = Kw + (size_t)NROW * NROW;
    float* v    = u + NROW;
    float* part = v + NROW;

    init_ones_kernel<<<NROW / 256, 256, 0, stream>>>(v);
    gemm_k_kernel<<<dim3(NROW / 128, NROW / 128), 256, 0, stream>>>(F, L, Kw);

    for (int s = 0; s < SINK_ITERS; ++s) {
        if ((s & 1) == 0) {
            // u = 1 / (K v)   : 13 times (s = 0,2,...,24)
            rowscale_kernel<<<NROW, 256, 0, stream>>>(Kw, v, u);
        } else {
            // v = 1 / (K^T u) : 12 times (s = 1,3,...,23)
            colpart_kernel<<<dim3(NROW / 64, 16), 256, 0, stream>>>(Kw, u, part);
            colfinish_kernel<<<NROW / 256, 256, 0, stream>>>(part, v);
        }
    }

    loss_kernel<<<1, 256, 0, stream>>>(Kw, u, v, out);
}